// DiagLRConv_23287312679212
// MI455X (gfx1250) — compile-verified
//
#include <hip/hip_runtime.h>

typedef float v2f __attribute__((ext_vector_type(2)));
typedef float v8f __attribute__((ext_vector_type(8)));

#define HW   262144   /* 512*512 */
#define WDIM 512
#define HDIM 512
#define CIN  16
#define COUT 16
#define KT   5

__global__ __launch_bounds__(256)
void diag_conv_wmma(const float* __restrict__ x,
                    const float* __restrict__ fw,
                    float* __restrict__ out) {
  const int lane = threadIdx.x & 31;
  const int wv   = threadIdx.x >> 5;   // wave in block: 0..7
  const int ph   = lane & 15;          // pixel-in-tile (B/C/D col) and A row M
  const int sel  = lane >> 4;          // half-wave select

  const int nh = blockIdx.x;           // n*512 + h
  const int n  = nh >> 9;
  const int h  = nh & 511;

  // ---- preload weights into A-operand registers ----
  // GEMM K index ordering: k = t*16 + i  (tap-major)
  // chunk c = t*4+cc covers k = 16t+4cc .. 16t+4cc+3
  // A 16x4 layout: lane L<16 holds (M=L, K=4c+{0,1}); L>=16 holds (M=L-16, K=4c+{2,3})
  v2f A[20];
  {
    const float* fwo = fw + ph * (CIN * KT);   // weight row o = ph
#pragma unroll
    for (int t = 0; t < KT; ++t) {
#pragma unroll
      for (int cc = 0; cc < 4; ++cc) {
        const int i0 = 4 * cc + 2 * sel;       // in-channel for this lane's K pair
        v2f a;
        a.x = fwo[i0 * KT + t];
        a.y = fwo[(i0 + 1) * KT + t];
        A[t * 4 + cc] = a;
      }
    }
  }

  const float* xn = x + (size_t)n * CIN * HW;  // image base (channel 0)

#pragma unroll 1
  for (int tt = 0; tt < 4; ++tt) {
    const int  w0    = wv * 64 + tt * 16;
    const bool w_int = (w0 >= 2) && (w0 + 17 <= WDIM - 1);

    v8f acc = {};

#pragma unroll
    for (int t = 0; t < KT; ++t) {
      const int d  = t - 2;
      const int hd = h + d;
      if ((unsigned)hd < (unsigned)HDIM) {     // wave-uniform: skip whole tap at image edge
        const float* rowp = xn + (size_t)hd * WDIM;  // channel-0 row for this tap
        const int wofs = w0 + ph + d;
        if (w_int) {
          // fast path: no per-lane bounds work
#pragma unroll
          for (int cc = 0; cc < 4; ++cc) {
            const int i0 = 4 * cc + 2 * sel;
            v2f b;
            b.x = rowp[(size_t)i0 * HW + wofs];
            b.y = rowp[(size_t)(i0 + 1) * HW + wofs];
            acc = __builtin_amdgcn_wmma_f32_16x16x4_f32(
                false, A[t * 4 + cc], false, b, (short)0, acc, false, false);
          }
        } else {
          // edge tiles: clamp address (always-valid load) + select-to-zero, EXEC stays full
          const bool v  = (unsigned)wofs < (unsigned)WDIM;
          const int  wo = v ? wofs : 0;
#pragma unroll
          for (int cc = 0; cc < 4; ++cc) {
            const int i0 = 4 * cc + 2 * sel;
            float b0 = rowp[(size_t)i0 * HW + wo];
            float b1 = rowp[(size_t)(i0 + 1) * HW + wo];
            v2f b;
            b.x = v ? b0 : 0.0f;
            b.y = v ? b1 : 0.0f;
            acc = __builtin_amdgcn_wmma_f32_16x16x4_f32(
                false, A[t * 4 + cc], false, b, (short)0, acc, false, false);
          }
        }
      }
    }

    // ---- store 16x16 f32 C/D tile ----
    // VGPR r: lanes 0-15 -> (M=r, N=lane); lanes 16-31 -> (M=r+8, N=lane-16)
    float* op = out + (size_t)n * COUT * HW + (size_t)h * WDIM + w0 + ph;
#pragma unroll
    for (int r = 0; r < 8; ++r) {
      const int M = r + 8 * sel;
      op[(size_t)M * HW] = acc[r];
    }
  }
}

extern "C" void kernel_launch(void* const* d_in, const int* in_sizes, int n_in,
                              void* d_out, int out_size, void* d_ws, size_t ws_size,
                              hipStream_t stream) {
  const float* x  = (const float*)d_in[0];   // (16,16,512,512) f32
  const float* fw = (const float*)d_in[1];   // (16,16,5) f32
  float* outp = (float*)d_out;               // (16,16,512,512) f32
  dim3 grid(16 * 512);                       // one block per (n, h) row
  dim3 block(256);                           // 8 waves, each covers 64 pixels
  hipLaunchKernelGGL(diag_conv_wmma, grid, block, 0, stream, x, fw, outp);
}